// CLAHE_31258771980541
// MI455X (gfx1250) — compile-verified
//
#include <hip/hip_runtime.h>
#include <hip/hip_bf16.h>
#include <math.h>

// ---------------- problem constants ----------------
#define BATCH     4
#define IMG_H     2048
#define IMG_W     2048
#define GRIDN     8
#define TILE_H    256              // IMG_H / GRIDN
#define TILE_W    256              // IMG_W / GRIDN
#define NBINS     256
#define NTILES    (BATCH * GRIDN * GRIDN)   // 256
#define TPIX      (TILE_H * TILE_W)         // 65536
#define CLIPV     640.0f           // int(2.5 * 65536 / 256)
#define NPIX      (BATCH * IMG_H * IMG_W)   // 16777216

typedef __attribute__((ext_vector_type(2))) float v2f;
typedef __attribute__((ext_vector_type(8))) float v8f;

// sortable-uint encoding of float for atomic min/max
__device__ __forceinline__ unsigned fenc(float f) {
    unsigned u = __float_as_uint(f);
    return (u & 0x80000000u) ? ~u : (u | 0x80000000u);
}
__device__ __forceinline__ float fdec(unsigned e) {
    unsigned u = (e & 0x80000000u) ? (e ^ 0x80000000u) : ~e;
    return __uint_as_float(u);
}

// ---------------- kernel 0: init workspace scalars ----------------
__global__ void clahe_init(unsigned* minmax) {
    if (threadIdx.x == 0) {
        minmax[0] = 0xFFFFFFFFu;  // encoded min identity
        minmax[1] = 0x00000000u;  // encoded max identity
    }
}

// ---------------- kernel 1: global min / max ----------------
__global__ void clahe_minmax(const float4* __restrict__ x, unsigned* minmax, int nvec) {
    __shared__ float smn[256];
    __shared__ float smx[256];
    int gid = blockIdx.x * blockDim.x + threadIdx.x;
    int stride = gridDim.x * blockDim.x;
    float mn = INFINITY, mx = -INFINITY;
    for (int i = gid; i < nvec; i += stride) {
        float4 v = x[i];
        mn = fminf(mn, fminf(fminf(v.x, v.y), fminf(v.z, v.w)));
        mx = fmaxf(mx, fmaxf(fmaxf(v.x, v.y), fmaxf(v.z, v.w)));
    }
    smn[threadIdx.x] = mn;
    smx[threadIdx.x] = mx;
    __syncthreads();
    for (int s = 128; s > 0; s >>= 1) {
        if (threadIdx.x < s) {
            smn[threadIdx.x] = fminf(smn[threadIdx.x], smn[threadIdx.x + s]);
            smx[threadIdx.x] = fmaxf(smx[threadIdx.x], smx[threadIdx.x + s]);
        }
        __syncthreads();
    }
    if (threadIdx.x == 0) {
        atomicMin(&minmax[0], fenc(smn[0]));
        atomicMax(&minmax[1], fenc(smx[0]));
    }
}

// ---------------- kernel 2: per-tile histograms (+ optional bin cache) ----
// one block (256 thr = 8 waves) per tile; 8 privatized LDS histograms
template <bool CACHE_BINS>
__global__ void clahe_hist(const float* __restrict__ x, const unsigned* __restrict__ minmax,
                           float* __restrict__ histF, unsigned char* __restrict__ bins) {
    __shared__ unsigned sh[8 * NBINS];
    const int t    = threadIdx.x;
    const int tile = blockIdx.x;                 // b*64 + gy*8 + gx
    const int b  = tile >> 6;
    const int gy = (tile >> 3) & 7;
    const int gx = tile & 7;
    const float mn  = fdec(minmax[0]);
    const float mx  = fdec(minmax[1]);
    const float inv = 1.0f / (mx - mn);

    for (int i = t; i < 8 * NBINS; i += 256) sh[i] = 0u;
    __syncthreads();

    const int wave = t >> 5;
    const size_t base = (size_t)b * (IMG_H * IMG_W) + (size_t)(gy * TILE_H) * IMG_W + gx * TILE_W;
    for (int r = 0; r < TILE_H; ++r) {
        const size_t off = base + (size_t)r * IMG_W + t;
        if (r + 1 < TILE_H)
            __builtin_prefetch(x + off + IMG_W, 0, 3);  // global_prefetch_b8
        float v  = x[off];
        float xn = (v - mn) * inv;
        int bin  = (int)(xn * 256.0f);
        bin = bin < 0 ? 0 : (bin > 255 ? 255 : bin);
        if (CACHE_BINS) bins[off] = (unsigned char)bin;
        atomicAdd(&sh[wave * NBINS + bin], 1u);
    }
    __syncthreads();

    unsigned total = 0;
    for (int w = 0; w < 8; ++w) total += sh[w * NBINS + t];
    histF[tile * NBINS + t] = (float)total;
}

// ---------------- kernel 3: clip + redistribute + WMMA cumsum -> LUT ----------------
// one wave32 per tile. cumsum(256) = H(16x16) x U  +  Lstrict x RS, via
// chained V_WMMA_F32_16X16X4_F32 (K accumulated in 4-chunks).
__global__ void clahe_lut(const float* __restrict__ histF, float* __restrict__ lut) {
    __shared__ float hc[NBINS];
    __shared__ float partial[32];
    __shared__ float rs[16];

    const int lane = threadIdx.x;        // 0..31
    const int tile = blockIdx.x;
    const int half = lane >> 4;          // 0 or 1
    const int M    = lane & 15;          // A row
    const int N    = lane & 15;          // B/C/D column

    // load + clip, accumulate partial sums (lane owns bins lane + 32*j)
    float psum = 0.0f;
    for (int j = 0; j < 8; ++j) {
        int i = lane + 32 * j;
        float h = fminf(histF[tile * NBINS + i], CLIPV);
        hc[i] = h;
        psum += h;
    }
    partial[lane] = psum;
    __syncthreads();

    float total = 0.0f;
    for (int i = 0; i < 32; ++i) total += partial[i];
    const float add = ((float)TPIX - total) * (1.0f / (float)NBINS);
    for (int j = 0; j < 8; ++j) hc[lane + 32 * j] += add;
    __syncthreads();

    // row sums of H (16 rows), lanes 0..15 each do one
    if (lane < 16) {
        float s = 0.0f;
        for (int c = 0; c < 16; ++c) s += hc[16 * lane + c];
        rs[lane] = s;
    }
    __syncthreads();

    v8f acc = {};   // C starts at zero

    // ---- acc += H x U  (within-row inclusive prefix) ----
    #pragma unroll
    for (int q = 0; q < 4; ++q) {
        const int k0 = 4 * q + 2 * half;
        const int k1 = k0 + 1;
        v2f a, bb;
        a[0]  = hc[16 * M + k0];
        a[1]  = hc[16 * M + k1];
        bb[0] = (k0 <= N) ? 1.0f : 0.0f;   // U[k][n] = k<=n
        bb[1] = (k1 <= N) ? 1.0f : 0.0f;
        acc = __builtin_amdgcn_wmma_f32_16x16x4_f32(false, a, false, bb,
                                                    (short)0, acc, false, false);
    }
    // ---- acc += Lstrict x RS  (sum of fully-completed rows) ----
    #pragma unroll
    for (int q = 0; q < 4; ++q) {
        const int k0 = 4 * q + 2 * half;
        const int k1 = k0 + 1;
        v2f a, bb;
        a[0]  = (k0 < M) ? 1.0f : 0.0f;    // Lstrict[m][k] = k<m
        a[1]  = (k1 < M) ? 1.0f : 0.0f;
        bb[0] = rs[k0];                    // RS[k][n] = rowsum(k)
        bb[1] = rs[k1];
        acc = __builtin_amdgcn_wmma_f32_16x16x4_f32(false, a, false, bb,
                                                    (short)0, acc, false, false);
    }

    // acc[v] = cumsum[row = v + 8*half][col = N]  ->  bin = 16*row + col
    #pragma unroll
    for (int v = 0; v < 8; ++v) {
        const int r = v + 8 * half;
        float cs = acc[v];
        float tq = cs * (255.0f / (float)TPIX);
        tq = fminf(fmaxf(tq, 0.0f), 255.0f);
        lut[tile * NBINS + 16 * r + N] = floorf(tq) * (1.0f / 255.0f);
    }
}

// ---------------- shared bilinear helper ----------------
__device__ __forceinline__ void bilin_coord(int h, float& wy, int& y0, int& y1) {
    float yc = (h + 0.5f) * (1.0f / TILE_H) - 0.5f;
    yc = fminf(fmaxf(yc, 0.0f), (float)(GRIDN - 1));
    y0 = (int)floorf(yc);
    y0 = y0 < 0 ? 0 : (y0 > GRIDN - 2 ? GRIDN - 2 : y0);
    y1 = y0 + 1;
    wy = yc - (float)y0;
}

__device__ __forceinline__ float blend(const float* __restrict__ lut, int b, int bin,
                                       int y0, int y1, float wy, int x0, int x1, float wx) {
    const int tb = b * 64;
    const float l00 = lut[((tb + y0 * 8 + x0) << 8) + bin];
    const float l01 = lut[((tb + y0 * 8 + x1) << 8) + bin];
    const float l10 = lut[((tb + y1 * 8 + x0) << 8) + bin];
    const float l11 = lut[((tb + y1 * 8 + x1) << 8) + bin];
    return (1.0f - wy) * ((1.0f - wx) * l00 + wx * l01)
         +         wy  * ((1.0f - wx) * l10 + wx * l11);
}

// ---------------- kernel 4a: bilinear apply from cached 8-bit bins ----------------
__global__ void clahe_apply_bins(const unsigned char* __restrict__ bins,
                                 const float* __restrict__ lut, float* __restrict__ out, int nvec) {
    int vid = blockIdx.x * blockDim.x + threadIdx.x;
    if (vid >= nvec) return;

    const int lin0 = vid * 4;
    const int b = lin0 / (IMG_H * IMG_W);
    const int rem = lin0 - b * (IMG_H * IMG_W);
    const int h = rem / IMG_W;
    const int w0 = rem - h * IMG_W;     // 4 consecutive pixels share b,h

    float wy; int y0, y1;
    bilin_coord(h, wy, y0, y1);

    const uchar4 bv = *(const uchar4*)(bins + lin0);
    const int pb[4] = {bv.x, bv.y, bv.z, bv.w};
    float po[4];

    #pragma unroll
    for (int j = 0; j < 4; ++j) {
        const int w = w0 + j;
        float xc = (w + 0.5f) * (1.0f / TILE_W) - 0.5f;
        xc = fminf(fmaxf(xc, 0.0f), (float)(GRIDN - 1));
        int x0 = (int)floorf(xc);
        x0 = x0 < 0 ? 0 : (x0 > GRIDN - 2 ? GRIDN - 2 : x0);
        const float wx = xc - (float)x0;
        po[j] = blend(lut, b, pb[j], y0, y1, wy, x0, x0 + 1, wx);
    }
    float4 vo = {po[0], po[1], po[2], po[3]};
    *(float4*)(out + lin0) = vo;
}

// ---------------- kernel 4b: bilinear apply, recompute bins from x (fallback) ----
__global__ void clahe_apply_recompute(const float* __restrict__ x, const unsigned* __restrict__ minmax,
                                      const float* __restrict__ lut, float* __restrict__ out, int nvec) {
    int vid = blockIdx.x * blockDim.x + threadIdx.x;
    if (vid >= nvec) return;

    const float mn  = fdec(minmax[0]);
    const float mx  = fdec(minmax[1]);
    const float inv = 1.0f / (mx - mn);

    const int lin0 = vid * 4;
    const int b = lin0 / (IMG_H * IMG_W);
    const int rem = lin0 - b * (IMG_H * IMG_W);
    const int h = rem / IMG_W;
    const int w0 = rem - h * IMG_W;

    float wy; int y0, y1;
    bilin_coord(h, wy, y0, y1);

    const float4 vx = *(const float4*)(x + lin0);
    const float px[4] = {vx.x, vx.y, vx.z, vx.w};
    float po[4];

    #pragma unroll
    for (int j = 0; j < 4; ++j) {
        const int w = w0 + j;
        float xc = (w + 0.5f) * (1.0f / TILE_W) - 0.5f;
        xc = fminf(fmaxf(xc, 0.0f), (float)(GRIDN - 1));
        int x0 = (int)floorf(xc);
        x0 = x0 < 0 ? 0 : (x0 > GRIDN - 2 ? GRIDN - 2 : x0);
        const float wx = xc - (float)x0;

        float xn = (px[j] - mn) * inv;
        int bin = (int)(xn * 256.0f);
        bin = bin < 0 ? 0 : (bin > 255 ? 255 : bin);
        po[j] = blend(lut, b, bin, y0, y1, wy, x0, x0 + 1, wx);
    }
    float4 vo = {po[0], po[1], po[2], po[3]};
    *(float4*)(out + lin0) = vo;
}

// ---------------- launcher ----------------
extern "C" void kernel_launch(void* const* d_in, const int* in_sizes, int n_in,
                              void* d_out, int out_size, void* d_ws, size_t ws_size,
                              hipStream_t stream) {
    const float* x = (const float*)d_in[0];
    float* out = (float*)d_out;

    char* ws = (char*)d_ws;
    unsigned* minmax = (unsigned*)ws;                            // 2 u32 (256B reserved)
    float* histF     = (float*)(ws + 256);                       // 256*256 f32
    float* lut       = (float*)(ws + 256 + NTILES * NBINS * 4);  // 256*256 f32
    unsigned char* bins = (unsigned char*)(ws + 256 + 2 * NTILES * NBINS * 4);

    const size_t ws_need_bins = 256 + 2 * (size_t)NTILES * NBINS * 4 + (size_t)NPIX;
    const bool use_bins = (ws_size >= ws_need_bins);

    clahe_init<<<1, 32, 0, stream>>>(minmax);
    clahe_minmax<<<1024, 256, 0, stream>>>((const float4*)x, minmax, NPIX / 4);
    if (use_bins) {
        clahe_hist<true><<<NTILES, 256, 0, stream>>>(x, minmax, histF, bins);
        clahe_lut<<<NTILES, 32, 0, stream>>>(histF, lut);
        clahe_apply_bins<<<(NPIX / 4 + 255) / 256, 256, 0, stream>>>(bins, lut, out, NPIX / 4);
    } else {
        clahe_hist<false><<<NTILES, 256, 0, stream>>>(x, minmax, histF, nullptr);
        clahe_lut<<<NTILES, 32, 0, stream>>>(histF, lut);
        clahe_apply_recompute<<<(NPIX / 4 + 255) / 256, 256, 0, stream>>>(x, minmax, lut, out, NPIX / 4);
    }
}